// histogram_binning_calibration_18957985644690
// MI455X (gfx1250) — compile-verified
//
#include <hip/hip_runtime.h>
#include <math.h>
#include <stdint.h>

#define TPB            256
#define NCLS           1000
#define V4T            250      // threads carrying a float4 (250*4 = 1000)
#define NBINS          15
#define ROWS_PER_BLOCK 8

// ---- CDNA5 async global->LDS copy (ASYNCcnt path) -------------------------
// VDST = LDS byte offset (low 32 bits of a generic __shared__ pointer are the
// group-segment offset on AMDGPU), VADDR = 64-bit global address, SADDR = off.
__device__ __forceinline__ void async_copy_b128(uint32_t lds_addr, const void* g) {
  asm volatile("global_load_async_to_lds_b128 %0, %1, off"
               :: "v"(lds_addr), "v"((uint64_t)(uintptr_t)g)
               : "memory");
}

template <int N>
__device__ __forceinline__ void wait_async_le() {
#if defined(__has_builtin) && __has_builtin(__builtin_amdgcn_s_wait_asynccnt)
  __builtin_amdgcn_s_wait_asynccnt(N);
#else
  asm volatile("s_wait_asynccnt %0" :: "n"(N) : "memory");
#endif
}

__global__ __launch_bounds__(TPB)
void hist_calib_kernel(const float* __restrict__ logits,
                       const float* __restrict__ hist,
                       float* __restrict__ out_sm,
                       float* __restrict__ out_valid,
                       float* __restrict__ out_pred,
                       int nrows) {
  __shared__ __align__(16) float buf[2][1024];   // two row buffers (double buffer)
  __shared__ float red_f[8];
  __shared__ int   red_i[8];
  __shared__ float bm_s;
  __shared__ int   bi_s;
  __shared__ float bc[3];                        // est, scale, invZ

  const int t    = threadIdx.x;
  const int wave = t >> 5;
  const int lane = t & 31;
  const int row0 = (int)blockIdx.x * ROWS_PER_BLOCK;
  const int rend = min(row0 + ROWS_PER_BLOCK, nrows);
  if (row0 >= nrows) return;

  // Prefetch first row of this block into buffer 0.
  if (t < V4T)
    async_copy_b128((uint32_t)(uintptr_t)&buf[0][t * 4],
                    logits + (size_t)row0 * NCLS + (size_t)t * 4);

  for (int r = 0; r < ROWS_PER_BLOCK; ++r) {
    const int row = row0 + r;
    if (row >= rend) break;

    const bool havenext = (row + 1 < rend);
    if (havenext && t < V4T)
      async_copy_b128((uint32_t)(uintptr_t)&buf[(r + 1) & 1][t * 4],
                      logits + (size_t)(row + 1) * NCLS + (size_t)t * 4);

    // Oldest outstanding async op (current row) must be complete.
    if (havenext) wait_async_le<1>(); else wait_async_le<0>();
    __syncthreads();

    const float* p = buf[r & 1];
    float4 v = make_float4(-INFINITY, -INFINITY, -INFINITY, -INFINITY);
    if (t < V4T) v = *(const float4*)(p + t * 4);

    // Local argmax, first-occurrence tie-break via strict '>'.
    float m = v.x; int mi = t * 4;
    if (v.y > m) { m = v.y; mi = t * 4 + 1; }
    if (v.z > m) { m = v.z; mi = t * 4 + 2; }
    if (v.w > m) { m = v.w; mi = t * 4 + 3; }
    if (t >= V4T) { m = -INFINITY; mi = 0x7fffffff; }

    // Wave32 (val, lowest-index) reduction.
    for (int off = 16; off; off >>= 1) {
      float om = __shfl_xor(m, off, 32);
      int   oi = __shfl_xor(mi, off, 32);
      if (om > m || (om == m && oi < mi)) { m = om; mi = oi; }
    }
    if (lane == 0) { red_f[wave] = m; red_i[wave] = mi; }
    __syncthreads();
    if (t == 0) {
      float bm = red_f[0]; int bidx = red_i[0];
      for (int w = 1; w < 8; ++w) {
        float om = red_f[w]; int oi = red_i[w];
        if (om > bm || (om == bm && oi < bidx)) { bm = om; bidx = oi; }
      }
      bm_s = bm; bi_s = bidx;
    }
    __syncthreads();
    const float bm = bm_s;
    const int pred = bi_s;

    // exp pass; keep e in registers for the store pass.
    float4 e;
    e.x = expf(v.x - bm); e.y = expf(v.y - bm);
    e.z = expf(v.z - bm); e.w = expf(v.w - bm);   // expf(-inf)=0 for idle lanes
    float s = (e.x + e.y) + (e.z + e.w);
    for (int off = 16; off; off >>= 1) s += __shfl_xor(s, off, 32);
    if (lane == 0) red_f[wave] = s;
    __syncthreads();
    if (t == 0) {
      float Z = 0.f;
      for (int w = 0; w < 8; ++w) Z += red_f[w];
      const float invZ = 1.0f / Z;
      const float conf = invZ;                    // max softmax = exp(0)/Z
      int bin = (int)ceilf(conf * (float)NBINS) - 1;
      bin = min(max(bin, 0), NBINS - 1);
      const float h     = hist[bin];
      const float valid = (h != -1.0f) ? 1.0f : 0.0f;
      const float est   = (h == -1.0f) ? conf : h;
      const float denom = (Z - 1.0f) * invZ;      // == sum of non-pred softmax
      const float scale = (1.0f - est) / denom;
      bc[0] = est; bc[1] = scale; bc[2] = invZ;
      out_valid[row] = valid;
      out_pred[row]  = (float)pred;
    }
    __syncthreads();

    const float est = bc[0], k = bc[2] * bc[1];   // invZ * (1-est)/denom
    if (t < V4T) {
      float4 o;
      o.x = e.x * k; o.y = e.y * k; o.z = e.z * k; o.w = e.w * k;
      const int base = t * 4;
      if (pred >= base && pred < base + 4) ((float*)&o)[pred - base] = est;
      *(float4*)(out_sm + (size_t)row * NCLS + base) = o;
    }
  }
}

extern "C" void kernel_launch(void* const* d_in, const int* in_sizes, int n_in,
                              void* d_out, int out_size, void* d_ws, size_t ws_size,
                              hipStream_t stream) {
  (void)n_in; (void)out_size; (void)d_ws; (void)ws_size;
  const float* logits = (const float*)d_in[0];
  const float* hist   = (const float*)d_in[1];
  const int nrows = in_sizes[0] / NCLS;

  float* out_sm    = (float*)d_out;
  float* out_valid = out_sm + (size_t)nrows * NCLS;
  float* out_pred  = out_valid + nrows;

  const int blocks = (nrows + ROWS_PER_BLOCK - 1) / ROWS_PER_BLOCK;
  hist_calib_kernel<<<blocks, TPB, 0, stream>>>(logits, hist, out_sm, out_valid,
                                                out_pred, nrows);
}